// ImprovedBilateralFilter3D_49658411877034
// MI455X (gfx1250) — compile-verified
//
#include <hip/hip_runtime.h>

// ---------------------------------------------------------------------------
// Fused gradient-guided 3D bilateral filter for 96^3 f32 volume (MI455X).
//  - one kernel: stage 12^3 tile -> LDS, Sobel grads (WMMA f32 16x16x4) -> LDS,
//    27-tap bilateral with a single v_exp_f32 per neighbor.
//  - zero-pad semantics for the Sobel conv, edge-clamp semantics for the
//    bilateral neighbor reads (matches the JAX reference exactly).
// ---------------------------------------------------------------------------

typedef __attribute__((ext_vector_type(2))) float v2f;
typedef __attribute__((ext_vector_type(8))) float v8f;

#define NVOL 96
#define TILE 8
#define SV   12            // staged tile dim (TILE + 4)
#define GV   10            // gradient tile dim (TILE + 2)
#define SV3  (SV*SV*SV)    // 1728
#define GV3  (GV*GV*GV)    // 1000
#define NTHREADS 512

__device__ __forceinline__ float smoothw(int i) { return (i == 1) ? 0.5f : 0.25f; }
__device__ __forceinline__ float derivw(int i)  { return (i == 0) ? 1.0f : ((i == 2) ? -1.0f : 0.0f); }

// Sobel filter-bank weight: row m (0..2 = d/dD, d/dH, d/dW; >=3 zero), tap t (0..26; >=27 zero pad)
__device__ __forceinline__ float sobel_w(int m, int t) {
    if (m >= 3 || t >= 27) return 0.0f;
    int di = t / 9, dj = (t / 3) % 3, dk = t % 3;
    float a = (m == 0) ? derivw(di) : smoothw(di);
    float b = (m == 1) ? derivw(dj) : smoothw(dj);
    float c = (m == 2) ? derivw(dk) : smoothw(dk);
    return a * b * c;
}

__device__ __forceinline__ int clampN(int v) {
    return v < 0 ? 0 : (v > NVOL - 1 ? NVOL - 1 : v);
}

__global__ void __launch_bounds__(NTHREADS)
bilateral3d_kernel(const float* __restrict__ vol, float* __restrict__ out) {
    __shared__ float svol[SV3];              // volume tile, ZERO padded outside [0,95]
    __shared__ float gH[GV3], gW[GV3], gD[GV3];  // gradient tile (global origin - 1)

    const int z0 = blockIdx.z * TILE;
    const int y0 = blockIdx.y * TILE;
    const int x0 = blockIdx.x * TILE;
    const int tid = threadIdx.x + TILE * threadIdx.y + TILE * TILE * threadIdx.z;

    // Hint the DMA path: pull this tile's region toward the caches early.
    __builtin_prefetch(vol + (((size_t)z0 * NVOL + y0) * NVOL + x0), 0, 3);

    // ---- stage 12^3 tile (zero fill outside the volume) -------------------
    for (int f = tid; f < SV3; f += NTHREADS) {
        int lz = f / (SV * SV);
        int ly = (f / SV) % SV;
        int lx = f % SV;
        int gz = z0 - 2 + lz, gy = y0 - 2 + ly, gx = x0 - 2 + lx;
        float v = 0.0f;
        if ((unsigned)gz < NVOL && (unsigned)gy < NVOL && (unsigned)gx < NVOL)
            v = vol[((size_t)gz * NVOL + gy) * NVOL + gx];
        svol[f] = v;
    }
    __syncthreads();

    // ---- Sobel gradients on the 10^3 halo region --------------------------
    const int lane = tid & 31;
    const int wave = tid >> 5;       // 16 waves
    const int half = lane >> 4;      // K-half selector for A/B operand layout
    const int mn   = lane & 15;      // A row m == B column n

#if __has_builtin(__builtin_amdgcn_wmma_f32_16x16x4_f32)
    // D(16x16) = A(16x4) x B(4x16) + C, accumulated over 7 K-steps (28 taps, 27 real).
    // A rows 0..2 = the three Sobel filters; B columns = 16 gradient points.
    for (int g = wave; g < (GV3 + 15) / 16; g += 16) {
        int p = g * 16 + mn;
        if (p > GV3 - 1) p = GV3 - 1;                 // pad lanes recompute last point
        int pz = p / (GV * GV), py = (p / GV) % GV, px = p % GV;
        int sbase = (pz * SV + py) * SV + px;         // svol coord of tap (0,0,0)

        v8f acc = {};
        #pragma unroll
        for (int kk = 0; kk < 7; ++kk) {
            int tA0 = 4 * kk + 2 * half;              // VGPR0 carries K+0 / K+2
            int tA1 = tA0 + 1;                        // VGPR1 carries K+1 / K+3
            v2f a, b;
            a.x = sobel_w(mn, tA0);
            a.y = sobel_w(mn, tA1);
            int t0 = tA0 > 26 ? 26 : tA0;             // pad tap: weight already 0
            int t1 = tA1 > 26 ? 26 : tA1;
            int o0 = ((t0 / 9) * SV + ((t0 / 3) % 3)) * SV + (t0 % 3);
            int o1 = ((t1 / 9) * SV + ((t1 / 3) % 3)) * SV + (t1 % 3);
            b.x = svol[sbase + o0];
            b.y = svol[sbase + o1];
            acc = __builtin_amdgcn_wmma_f32_16x16x4_f32(
                false, a, false, b, (short)0, acc, false, false);
        }
        if (half == 0) {                              // lanes 0..15 hold rows M=0..7
            gH[p] = acc[0];                           // M=0: d/dD
            gW[p] = acc[1];                           // M=1: d/dH
            gD[p] = acc[2];                           // M=2: d/dW
        }
    }
#else
    // VALU fallback (identical math)
    for (int p = tid; p < GV3; p += NTHREADS) {
        int pz = p / (GV * GV), py = (p / GV) % GV, px = p % GV;
        int sbase = (pz * SV + py) * SV + px;
        float h = 0.f, w = 0.f, d = 0.f;
        #pragma unroll
        for (int t = 0; t < 27; ++t) {
            int o = ((t / 9) * SV + ((t / 3) % 3)) * SV + (t % 3);
            float v = svol[sbase + o];
            h += sobel_w(0, t) * v;
            w += sobel_w(1, t) * v;
            d += sobel_w(2, t) * v;
        }
        gH[p] = h; gW[p] = w; gD[p] = d;
    }
#endif
    __syncthreads();

    // ---- 27-neighbor bilateral (edge-clamped coords) ----------------------
    const int gz = z0 + threadIdx.z;
    const int gy = y0 + threadIdx.y;
    const int gx = x0 + threadIdx.x;

    const int cg = (((int)threadIdx.z + 1) * GV + ((int)threadIdx.y + 1)) * GV + ((int)threadIdx.x + 1);
    const float ch = gH[cg], cw = gW[cg], cd = gD[cg];

    float num = 0.0f, den = 0.0f;
    #pragma unroll
    for (int dz = -1; dz <= 1; ++dz)
    #pragma unroll
    for (int dy = -1; dy <= 1; ++dy)
    #pragma unroll
    for (int dx = -1; dx <= 1; ++dx) {
        int qz = clampN(gz + dz), qy = clampN(gy + dy), qx = clampN(gx + dx);
        int qg = ((qz - (z0 - 1)) * GV + (qy - (y0 - 1))) * GV + (qx - (x0 - 1));
        float eh = gH[qg] - ch;
        float ew = gW[qg] - cw;
        float ed = gD[qg] - cd;
        float r2 = eh * eh + ew * ew + ed * ed;
        float s2 = (float)(dz * dz + dy * dy + dx * dx);
        // sigma_r = 1.2 -> 2*sr^2 = 2.88 ; sigma_d = 120 -> 2*sd^2 = 28800
        float wgt = __expf(-(r2 * (1.0f / 2.88f) + s2 * (1.0f / 28800.0f)));
        float v = svol[((qz - (z0 - 2)) * SV + (qy - (y0 - 2))) * SV + (qx - (x0 - 2))];
        num += wgt * v;
        den += wgt;
    }

    out[((size_t)gz * NVOL + gy) * NVOL + gx] = num / fmaxf(den, 1e-8f);
}

extern "C" void kernel_launch(void* const* d_in, const int* in_sizes, int n_in,
                              void* d_out, int out_size, void* d_ws, size_t ws_size,
                              hipStream_t stream) {
    (void)in_sizes; (void)n_in; (void)out_size; (void)d_ws; (void)ws_size;
    const float* vol = (const float*)d_in[0];
    float* out = (float*)d_out;
    dim3 grid(NVOL / TILE, NVOL / TILE, NVOL / TILE);   // 12^3 workgroups
    dim3 block(TILE, TILE, TILE);                       // 512 threads = 16 wave32
    bilateral3d_kernel<<<grid, block, 0, stream>>>(vol, out);
}